// SoftmaxMatcher_40621800685629
// MI455X (gfx1250) — compile-verified
//
#include <hip/hip_runtime.h>

// ---------------------------------------------------------------------------
// Problem constants (from the reference)
// ---------------------------------------------------------------------------
#define N_   4096
#define M_   4096
#define D_   512
#define NP1  4097               // N+1 == M+1
#define LD   NP1                // Z row stride (Z lives in d_out)
#define LMBA 10.0f              // 1/REG
#define PHI_OVER_LMBA 0.00909090909090909f   // (REG_KL/(REG_KL+REG)) / lmba
#define LOG_MU_IN  (-8.42312670f)            // log(0.9/4096)
#define LOG_MU_BIN (-2.30258509f)            // log(0.1)
#define NSPLIT 32
#define ROWS_PER_SPLIT 129      // ceil(4097/32)
#define PPAD 4104               // padded partial stride

typedef __attribute__((ext_vector_type(2))) float v2f;
typedef __attribute__((ext_vector_type(8))) float v8f;

// ---------------------------------------------------------------------------
// 1) Per-row inverse L2 norm: one wave (32 lanes) per row, float4 loads
// ---------------------------------------------------------------------------
__global__ void row_invnorm_kernel(const float* __restrict__ ft,
                                   float* __restrict__ inv, int rows) {
    int wave = (blockIdx.x * blockDim.x + threadIdx.x) >> 5;
    int lane = threadIdx.x & 31;
    if (wave >= rows) return;
    const float* r = ft + (size_t)wave * D_;
    float s = 0.f;
    #pragma unroll
    for (int t = 0; t < D_ / 128; ++t) {
        float4 q = *(const float4*)(r + t * 128 + lane * 4);
        s += q.x * q.x + q.y * q.y + q.z * q.z + q.w * q.w;
    }
    #pragma unroll
    for (int off = 16; off > 0; off >>= 1) s += __shfl_xor(s, off, 32);
    if (lane == 0) inv[wave] = rsqrtf(s);
}

// ---------------------------------------------------------------------------
// 2) Dustbin row/col of Z (= lmba*bin_score) + zero-init u, v
// ---------------------------------------------------------------------------
__global__ void fill_dustbin_kernel(float* __restrict__ Z,
                                    const float* __restrict__ bin,
                                    float* __restrict__ u, float* __restrict__ v) {
    int idx = blockIdx.x * blockDim.x + threadIdx.x;
    if (idx >= NP1) return;
    float zb = LMBA * bin[0];
    Z[(size_t)N_ * LD + idx] = zb;   // last row
    Z[(size_t)idx * LD + M_] = zb;   // last col (corner written too; zeroed at end)
    u[idx] = 0.f;
    v[idx] = 0.f;
}

// ---------------------------------------------------------------------------
// 3) FP32 WMMA GEMM: Z[i,j] = lmba * inv0[i]*inv1[j] * (ft0_i . ft1_j)
//    One 16x16 C tile per wave; v_wmma_f32_16x16x4_f32 over K=512.
//    A frag (16x4, MxK): lane<16 -> {A[lm][k],A[lm][k+1]}, lane>=16 -> k+2,k+3
//    B frag (4x16, KxN): symmetric striping, B[k][n] = ft1[n][k]
// ---------------------------------------------------------------------------
__global__ void gemm_wmma_kernel(const float* __restrict__ A,
                                 const float* __restrict__ B,
                                 const float* __restrict__ inv0,
                                 const float* __restrict__ inv1,
                                 float* __restrict__ Z) {
    int wave = threadIdx.x >> 5;
    int lane = threadIdx.x & 31;
    int half = lane >> 4;        // 0: lanes 0-15, 1: lanes 16-31
    int lm   = lane & 15;
    int tm = (blockIdx.y * 8 + wave) * 16;   // C-tile row base
    int tn = blockIdx.x * 16;                // C-tile col base

    const float* arow = A + (size_t)(tm + lm) * D_ + 2 * half;
    const float* brow = B + (size_t)(tn + lm) * D_ + 2 * half;

    v8f c = {};
    #pragma unroll 8
    for (int k = 0; k < D_; k += 4) {
        v2f a = *(const v2f*)(arow + k);
        v2f b = *(const v2f*)(brow + k);
        c = __builtin_amdgcn_wmma_f32_16x16x4_f32(
                /*neg_a=*/false, a, /*neg_b=*/false, b,
                /*c_mod=*/(short)0, c, /*reuse_a=*/false, /*reuse_b=*/false);
    }

    // Epilogue: fold normalization + lmba; C layout: VGPR r -> row = r + 8*half
    int   col = tn + lm;
    float s1  = LMBA * inv1[col];
    int   rb  = tm + 8 * half;
    #pragma unroll
    for (int r = 0; r < 8; ++r) {
        Z[(size_t)(rb + r) * LD + col] = c[r] * s1 * inv0[rb + r];
    }
}

// ---------------------------------------------------------------------------
// 4) Sinkhorn row pass: u[i] = (log_mu[i] - LSE_j(Z[i,j] + v[j]*lmba)) * phi/lmba
//    One wave per row; online logsumexp; wave32 shfl reduction.
// ---------------------------------------------------------------------------
__global__ void row_pass_kernel(const float* __restrict__ Z,
                                const float* __restrict__ v,
                                float* __restrict__ u) {
    int i    = blockIdx.x * 8 + (threadIdx.x >> 5);
    int lane = threadIdx.x & 31;
    if (i >= NP1) return;
    const float* zr = Z + (size_t)i * LD;
    float m = -__builtin_inff(), s = 0.f;
    for (int j = lane; j < NP1; j += 32) {
        float x  = zr[j] + v[j] * LMBA;
        float mn = fmaxf(m, x);
        s = s * __expf(m - mn) + __expf(x - mn);
        m = mn;
    }
    #pragma unroll
    for (int off = 16; off > 0; off >>= 1) {
        float m2 = __shfl_xor(m, off, 32);
        float s2 = __shfl_xor(s, off, 32);
        float mn = fmaxf(m, m2);
        s = s * __expf(m - mn) + s2 * __expf(m2 - mn);
        m = mn;
    }
    if (lane == 0) {
        float lse = m + __logf(s);
        float lmu = (i < N_) ? LOG_MU_IN : LOG_MU_BIN;
        u[i] = (lmu - lse) * PHI_OVER_LMBA;
    }
}

// ---------------------------------------------------------------------------
// 5a) Column pass partials: thread-per-column, 32 row splits for occupancy
// ---------------------------------------------------------------------------
__global__ void col_partial_kernel(const float* __restrict__ Z,
                                   const float* __restrict__ u,
                                   float* __restrict__ pm,
                                   float* __restrict__ ps) {
    int j = blockIdx.x * blockDim.x + threadIdx.x;
    int split = blockIdx.y;
    if (j >= NP1) return;
    int i0 = split * ROWS_PER_SPLIT;
    int i1 = i0 + ROWS_PER_SPLIT; if (i1 > NP1) i1 = NP1;
    float m = -__builtin_inff(), s = 0.f;
    #pragma unroll 4
    for (int i = i0; i < i1; ++i) {
        float x  = Z[(size_t)i * LD + j] + u[i] * LMBA;
        float mn = fmaxf(m, x);
        s = s * __expf(m - mn) + __expf(x - mn);
        m = mn;
    }
    pm[split * PPAD + j] = m;
    ps[split * PPAD + j] = s;
}

// ---------------------------------------------------------------------------
// 5b) Combine partials: v[j] = (log_nu[j] - LSE_i(...)) * phi/lmba
// ---------------------------------------------------------------------------
__global__ void col_reduce_kernel(const float* __restrict__ pm,
                                  const float* __restrict__ ps,
                                  float* __restrict__ v) {
    int j = blockIdx.x * blockDim.x + threadIdx.x;
    if (j >= NP1) return;
    float m = -__builtin_inff(), s = 0.f;
    #pragma unroll
    for (int sp = 0; sp < NSPLIT; ++sp) {
        float m2 = pm[sp * PPAD + j];
        float s2 = ps[sp * PPAD + j];
        float mn = fmaxf(m, m2);
        s = s * __expf(m - mn) + s2 * __expf(m2 - mn);
        m = mn;
    }
    float lse = m + __logf(s);
    float lnu = (j < M_) ? LOG_MU_IN : LOG_MU_BIN;
    v[j] = (lnu - lse) * PHI_OVER_LMBA;
}

// ---------------------------------------------------------------------------
// 6) Final: prob = exp(Z + u*lmba + v*lmba), corner -> 0 (in place in d_out)
// ---------------------------------------------------------------------------
__global__ void final_kernel(float* __restrict__ Z,
                             const float* __restrict__ u,
                             const float* __restrict__ v) {
    int j = blockIdx.x * blockDim.x + threadIdx.x;
    int i = blockIdx.y;
    if (j >= NP1) return;
    size_t idx = (size_t)i * LD + j;
    float p = __expf(Z[idx] + u[i] * LMBA + v[j] * LMBA);
    if (i == N_ && j == M_) p = 0.f;
    Z[idx] = p;
}

// ---------------------------------------------------------------------------
// Launcher
// ---------------------------------------------------------------------------
extern "C" void kernel_launch(void* const* d_in, const int* in_sizes, int n_in,
                              void* d_out, int out_size, void* d_ws, size_t ws_size,
                              hipStream_t stream) {
    const float* ft0 = (const float*)d_in[0];
    const float* ft1 = (const float*)d_in[1];
    const float* bin = (const float*)d_in[2];
    float* Z = (float*)d_out;                 // 4097*4097 f32, used as Z then prob
    float* w = (float*)d_ws;

    float* inv0 = w;                          // 4096
    float* inv1 = w + 4096;                   // 4096
    float* u    = w + 8192;                   // 4104 (4097 used)
    float* v    = w + 8192 + PPAD;            // 4104
    float* pm   = w + 8192 + 2 * PPAD;        // 32*4104
    float* ps   = pm + NSPLIT * PPAD;         // 32*4104

    // 1) inverse norms (one wave/row, 8 rows per 256-thread block)
    row_invnorm_kernel<<<N_ / 8, 256, 0, stream>>>(ft0, inv0, N_);
    row_invnorm_kernel<<<M_ / 8, 256, 0, stream>>>(ft1, inv1, M_);

    // 2) dustbin row/col + u=v=0
    fill_dustbin_kernel<<<(NP1 + 255) / 256, 256, 0, stream>>>(Z, bin, u, v);

    // 3) FP32-WMMA GEMM into Z interior
    gemm_wmma_kernel<<<dim3(M_ / 16, N_ / 16 / 8), 256, 0, stream>>>(
        ft0, ft1, inv0, inv1, Z);

    // 4-5) 100 Sinkhorn iterations (Z stays L2-resident: 67 MB < 192 MB L2)
    const int row_blocks = (NP1 + 7) / 8;          // 513
    const int col_blocks = (NP1 + 255) / 256;      // 17
    for (int it = 0; it < 100; ++it) {
        row_pass_kernel<<<row_blocks, 256, 0, stream>>>(Z, v, u);
        col_partial_kernel<<<dim3(col_blocks, NSPLIT), 256, 0, stream>>>(Z, u, pm, ps);
        col_reduce_kernel<<<col_blocks, 256, 0, stream>>>(pm, ps, v);
    }

    // 6) exponentiate in place, zero the corner
    final_kernel<<<dim3(col_blocks, NP1), 256, 0, stream>>>(Z, u, v);
}